// AspectImportanceEstimation_43525198578426
// MI455X (gfx1250) — compile-verified
//
#include <hip/hip_runtime.h>

typedef float v2f __attribute__((ext_vector_type(2)));
typedef float v8f __attribute__((ext_vector_type(8)));

#define NWAVES 8
#define TILE_FLOATS 1792  // 3*256 (U,V,T/aff) + 2*512 (HU, HI 32x16)

__device__ __forceinline__ v8f wmma_f32(v2f a, v2f b, v8f c) {
  return __builtin_amdgcn_wmma_f32_16x16x4_f32(false, a, false, b, (short)0, c,
                                               false, false);
}

__launch_bounds__(256, 2)
__global__ void aspect_importance_kernel(
    const float* __restrict__ g_user, const float* __restrict__ g_item,
    const float* __restrict__ g_M,    const float* __restrict__ g_Pu,
    const float* __restrict__ g_pu,   const float* __restrict__ g_Pi,
    const float* __restrict__ g_pi,   float* __restrict__ g_out, int nbatch)
{
  __shared__ float lds[NWAVES * TILE_FLOATS];
  const int tid  = threadIdx.x;
  const int wave = tid >> 5;
  const int lane = tid & 31;
  const int half = lane >> 4;   // which 16-lane half of the wave
  const int l15  = lane & 15;

  float* tU  = lds + wave * TILE_FLOATS; // 16x16 user tile (zero padded)
  float* tV  = tU + 256;                 // 16x16 item tile
  float* tT  = tV + 256;                 // T = U@M, then relu(aff)
  float* tHU = tT + 256;                 // 32x16 h_user staging
  float* tHI = tHU + 512;                // 32x16 h_item staging

  // zero the pad regions once; per-batch writes only touch [0..9][0..9]
  for (int i = lane; i < 256; i += 32) { tU[i] = 0.f; tV[i] = 0.f; }

  // ---- per-wave constant fragments ----
  // B-fragment of affinity matrix M (K chunks of 4): B[k][n] = M[k][n]
  v2f bM[3];
  #pragma unroll
  for (int c = 0; c < 3; ++c) {
    int k0 = 4*c + 2*half;
    float x = (k0     < 10 && l15 < 10) ? g_M[(k0    )*10 + l15] : 0.f;
    float y = (k0 + 1 < 10 && l15 < 10) ? g_M[(k0 + 1)*10 + l15] : 0.f;
    bM[c] = (v2f){x, y};
  }
  // A-fragments of the 32x10 projections: two 16-row tiles, K = d
  v2f aPu[2][3], aPi[2][3];
  #pragma unroll
  for (int t = 0; t < 2; ++t)
    #pragma unroll
    for (int c = 0; c < 3; ++c) {
      int row = l15 + 16*t;
      int d0  = 4*c + 2*half;
      float ux = (d0     < 10) ? g_Pu[row*10 + d0    ] : 0.f;
      float uy = (d0 + 1 < 10) ? g_Pu[row*10 + d0 + 1] : 0.f;
      float ix = (d0     < 10) ? g_Pi[row*10 + d0    ] : 0.f;
      float iy = (d0 + 1 < 10) ? g_Pi[row*10 + d0 + 1] : 0.f;
      aPu[t][c] = (v2f){ux, uy};
      aPi[t][c] = (v2f){ix, iy};
    }
  // per-lane slice of parameter vectors: k = r + 8*half + 16*t (D-frag rows)
  float puv[16], piv[16];
  #pragma unroll
  for (int t = 0; t < 2; ++t)
    #pragma unroll
    for (int r = 0; r < 8; ++r) {
      int k = r + 8*half + 16*t;
      puv[t*8 + r] = g_pu[k];
      piv[t*8 + r] = g_pi[k];
    }

  const int waveGlobal = blockIdx.x * NWAVES + wave;
  const int waveCount  = gridDim.x * NWAVES;
  float* outU = g_out;
  float* outI = g_out + (size_t)nbatch * 10;

  for (int b = waveGlobal; b < nbatch; b += waveCount) {
    // ---- stage U, V (10x10) into zero-padded tiles ----
    const float* srcU = g_user + (size_t)b * 100;
    const float* srcI = g_item + (size_t)b * 100;
    for (int i = lane; i < 100; i += 32) {
      int r = i / 10, cc = i - r * 10;
      tU[r*16 + cc] = __builtin_nontemporal_load(srcU + i);
      tV[r*16 + cc] = __builtin_nontemporal_load(srcI + i);
    }

    // ---- T = U @ M ----
    v8f acc = {};
    #pragma unroll
    for (int c = 0; c < 3; ++c) {
      int k0 = 4*c + 2*half;
      v2f a = (v2f){ tU[l15*16 + k0], tU[l15*16 + k0 + 1] };
      acc = wmma_f32(a, bM[c], acc);
    }
    #pragma unroll
    for (int r = 0; r < 8; ++r) tT[(r + 8*half)*16 + l15] = acc[r];

    // B-fragments of U^T and V^T (transposed LDS reads); bVT reused for h_item
    v2f bUT[3], bVT[3];
    #pragma unroll
    for (int c = 0; c < 3; ++c) {
      int k0 = 4*c + 2*half;
      bUT[c] = (v2f){ tU[l15*16 + k0], tU[l15*16 + k0 + 1] };
      bVT[c] = (v2f){ tV[l15*16 + k0], tV[l15*16 + k0 + 1] };
    }

    // ---- aff = relu(T @ V^T) ----
    v8f affd = {};
    #pragma unroll
    for (int c = 0; c < 3; ++c) {
      int k0 = 4*c + 2*half;
      v2f a = (v2f){ tT[l15*16 + k0], tT[l15*16 + k0 + 1] };
      affd = wmma_f32(a, bVT[c], affd);
    }
    #pragma unroll
    for (int r = 0; r < 8; ++r) affd[r] = fmaxf(affd[r], 0.f);
    #pragma unroll
    for (int r = 0; r < 8; ++r) tT[(r + 8*half)*16 + l15] = affd[r];

    // ---- h_user = Pu @ U^T ; h_item = Pi @ V^T ----
    v8f hU[2], hI[2];
    #pragma unroll
    for (int t = 0; t < 2; ++t) {
      v8f au = {}, ai = {};
      #pragma unroll
      for (int c = 0; c < 3; ++c) {
        au = wmma_f32(aPu[t][c], bUT[c], au);
        ai = wmma_f32(aPi[t][c], bVT[c], ai);
      }
      hU[t] = au; hI[t] = ai;
      #pragma unroll
      for (int r = 0; r < 8; ++r) {
        tHU[(r + 8*half + 16*t)*16 + l15] = au[r];
        tHI[(r + 8*half + 16*t)*16 + l15] = ai[r];
      }
    }

    // B-fragments of aff (direct) and aff^T (transposed) from the aff tile
    v2f bA[3], bAT[3];
    #pragma unroll
    for (int c = 0; c < 3; ++c) {
      int k0 = 4*c + 2*half;
      bA[c]  = (v2f){ tT[(k0)*16 + l15], tT[(k0 + 1)*16 + l15] };
      bAT[c] = (v2f){ tT[l15*16 + k0],   tT[l15*16 + k0 + 1]   };
    }

    // ---- hu = relu(h_user + h_item @ aff^T) ; u_score = p_u . hu ----
    float su = 0.f, si = 0.f;
    #pragma unroll
    for (int t = 0; t < 2; ++t) {
      v8f a = hU[t];
      #pragma unroll
      for (int c = 0; c < 3; ++c) {
        int k0 = 4*c + 2*half;
        v2f af = (v2f){ tHI[(l15 + 16*t)*16 + k0], tHI[(l15 + 16*t)*16 + k0 + 1] };
        a = wmma_f32(af, bAT[c], a);
      }
      #pragma unroll
      for (int r = 0; r < 8; ++r) su += puv[t*8 + r] * fmaxf(a[r], 0.f);
    }
    // ---- hi = relu(h_item + h_user @ aff) ; i_score = p_i . hi ----
    #pragma unroll
    for (int t = 0; t < 2; ++t) {
      v8f a = hI[t];
      #pragma unroll
      for (int c = 0; c < 3; ++c) {
        int k0 = 4*c + 2*half;
        v2f af = (v2f){ tHU[(l15 + 16*t)*16 + k0], tHU[(l15 + 16*t)*16 + k0 + 1] };
        a = wmma_f32(af, bA[c], a);
      }
      #pragma unroll
      for (int r = 0; r < 8; ++r) si += piv[t*8 + r] * fmaxf(a[r], 0.f);
    }
    // combine the two 16-lane halves (rows k split across halves, same column n)
    su += __shfl_xor(su, 16);
    si += __shfl_xor(si, 16);

    // ---- softmax over aspects n = l15 (valid n < 10), both halves duplicate ----
    bool valid = (l15 < 10);
    float mu = valid ? su : -__builtin_inff();
    float mi = valid ? si : -__builtin_inff();
    #pragma unroll
    for (int m = 1; m < 16; m <<= 1) {
      mu = fmaxf(mu, __shfl_xor(mu, m));
      mi = fmaxf(mi, __shfl_xor(mi, m));
    }
    float eu = valid ? __expf(su - mu) : 0.f;
    float ei = valid ? __expf(si - mi) : 0.f;
    float du = eu, di = ei;
    #pragma unroll
    for (int m = 1; m < 16; m <<= 1) {
      du += __shfl_xor(du, m);
      di += __shfl_xor(di, m);
    }
    if (half == 0 && valid) {
      __builtin_nontemporal_store(eu / du, outU + (size_t)b * 10 + l15);
      __builtin_nontemporal_store(ei / di, outI + (size_t)b * 10 + l15);
    }
  }
}

extern "C" void kernel_launch(void* const* d_in, const int* in_sizes, int n_in,
                              void* d_out, int out_size, void* d_ws, size_t ws_size,
                              hipStream_t stream) {
  const float* user_rep = (const float*)d_in[0];
  const float* item_rep = (const float*)d_in[1];
  const float* affm     = (const float*)d_in[2];
  const float* uproj    = (const float*)d_in[3];
  const float* upar     = (const float*)d_in[4];
  const float* iproj    = (const float*)d_in[5];
  const float* ipar     = (const float*)d_in[6];
  float* out = (float*)d_out;

  int nbatch = in_sizes[0] / 100;  // B*A*D / (A*D)
  int blocks = (nbatch + NWAVES - 1) / NWAVES;
  if (blocks > 2048) blocks = 2048;  // persistent grid, grid-stride over batch
  if (blocks < 1) blocks = 1;

  hipLaunchKernelGGL(aspect_importance_kernel, dim3(blocks), dim3(256), 0, stream,
                     user_rep, item_rep, affm, uproj, upar, iproj, ipar, out,
                     nbatch);
}